// DetectionLoss_71459665871196
// MI455X (gfx1250) — compile-verified
//
#include <hip/hip_runtime.h>
#include <hip/hip_bf16.h>
#include <stdint.h>

// ---------------- problem constants (match reference) ----------------
namespace {
constexpr int   kB      = 32;
constexpr int   kP      = 262144;
constexpr int   kNumNeg = 10000;
constexpr int   kBlkPerSample = 64;     // 4096 elements per block (256 thr x 16)
constexpr int   kHistBins = 65536;      // 16-bit radix phase
}

// AS-qualified typedefs for the async global->LDS path.
// (round-1 diagnostic confirmed param0 is addrspace(1) int*, param1 addrspace(3) int*)
typedef __attribute__((address_space(1))) int gbl_i32_t;
typedef __attribute__((address_space(3))) int lds_i32_t;

#if __has_builtin(__builtin_amdgcn_global_load_async_to_lds_b32)
#define HAVE_ASYNC_LDS 1
#endif

// WMMA vector types (wave32: A 16x4 f32 = 2 VGPRs, C/D 16x16 f32 = 8 VGPRs)
typedef float v2f __attribute__((ext_vector_type(2)));
typedef float v8f __attribute__((ext_vector_type(8)));

#if __has_builtin(__builtin_amdgcn_wmma_f32_16x16x4_f32)
#define HAVE_WMMA_F32 1
#endif

// ---------------- element loss (matches reference math) ----------------
__device__ __forceinline__ float loss_elem(float p, float t, float m,
                                           bool& pos, bool& neg) {
  pos = (t == 1.0f);
  neg = (t == 0.0f);
  float prob = 1.0f / (1.0f + expf(-p));
  prob = fminf(fmaxf(prob, 1.0e-4f), 1.0f - 1.0e-4f);
  const float af = pos ? 0.75f : 0.25f;                 // ALPHA
  const float w  = pos ? (1.0f - prob) : prob;
  const float focal = af * w * w;                        // GAMMA = 2
  // softplus(p) - p*t, numerically stable
  const float bce = fmaxf(p, 0.0f) + log1pf(expf(-fabsf(p))) - p * t;
  float loss = (m == 0.0f) ? focal * bce : 0.0f;
  if (pos && prob < 0.8f) loss *= 4.0f;                  // FN_WEIGHT
  const float hfp = 1.5f + fminf(fmaxf((prob - 0.5f) * 5.0f, 0.0f), 1.0f) * 0.5f;
  if (neg && prob > 0.5f) loss *= hfp;                   // HFP
  return loss;
}

// ---------------- WMMA-based exact wave32 sum ----------------
// D = A x ones + 0 with A-VGPR0 = lane value, A-VGPR1 = 0.
// ISA A layout (16x4 f32): lane l -> M=l%16, K=(l<16? {0,1} : {2,3}) for VGPR{0,1}
// => D[m][n] = a_m + a_(m+16) for every n (exact f32 adds, RNE).
// C/D layout: VGPR r at lane l -> M = r + 8*(l/16); summing the 8 D regs gives the
// half total; shfl_xor(16) adds the other half -> full 32-lane sum in every lane.
__device__ __forceinline__ float wave_sum_32(float v) {
#ifdef HAVE_WMMA_F32
  v2f a;  a.x = v;     a.y = 0.0f;
  v2f bo; bo.x = 1.0f; bo.y = 1.0f;
  v8f c = {};
  v8f d = __builtin_amdgcn_wmma_f32_16x16x4_f32(
      /*neg_a=*/false, a, /*neg_b=*/false, bo,
      /*c_mod=*/(short)0, c, /*reuse_a=*/false, /*reuse_b=*/false);
  float s = ((d[0] + d[1]) + (d[2] + d[3])) + ((d[4] + d[5]) + (d[6] + d[7]));
  s += __shfl_xor(s, 16, 32);
  return s;
#else
  for (int off = 16; off > 0; off >>= 1) v += __shfl_xor(v, off, 32);
  return v;
#endif
}

// Block (1024 thr = 32 waves) reduction: wave-level WMMA sums, then wave 0
// reduces the 32 per-wave partials with one more WMMA wave sum.
__device__ __forceinline__ float block_reduce_1024(float v, float* sred, int tid) {
  const int wave = tid >> 5;
  const int lane = tid & 31;
  const float ws = wave_sum_32(v);           // all lanes active at call sites
  if (lane == 0) sred[wave] = ws;
  __syncthreads();
  if (wave == 0) {                           // wave-uniform branch: EXEC all-1s
    const float r = wave_sum_32(sred[lane]);
    if (lane == 0) sred[0] = r;
  }
  __syncthreads();
  const float r = sred[0];
  __syncthreads();
  return r;
}

// ---------------- K0: zero metadata region ----------------
__global__ void det_zero_kernel(uint32_t* __restrict__ p, int n) {
  int i = blockIdx.x * 256 + threadIdx.x;
  if (i < n) p[i] = 0u;
}

// ---------------- K1: fused main pass ----------------
// loss stats + per-sample 16-bit-high histogram of neg_rand over negatives
__global__ __launch_bounds__(256)
void det_main_kernel(const float* __restrict__ pred,
                     const float* __restrict__ target,
                     const float* __restrict__ mask,
                     const float* __restrict__ rnd,
                     uint32_t* __restrict__ hist,
                     uint32_t* __restrict__ numPos,
                     uint32_t* __restrict__ numNeg,
                     float*    __restrict__ posPartials) {
  const int b   = blockIdx.y;
  const int tid = threadIdx.x;
  const size_t sbase = (size_t)b * kP;
  const int chunk = blockIdx.x * 4096;
  uint32_t* __restrict__ h = hist + (size_t)b * kHistBins;

  float posSum = 0.0f;
  uint32_t posCnt = 0, negCnt = 0;

#pragma unroll
  for (int j = 0; j < 4; ++j) {
    const size_t off = sbase + (size_t)chunk + (size_t)j * 1024 + (size_t)tid * 4;
    const float4 p4 = *(const float4*)(pred   + off);
    const float4 t4 = *(const float4*)(target + off);
    const float4 m4 = *(const float4*)(mask   + off);
    const float4 r4 = *(const float4*)(rnd    + off);
    if (j < 3) {                      // gfx1250: global_prefetch_b8
      __builtin_prefetch(pred   + off + 1024, 0, 0);
      __builtin_prefetch(rnd    + off + 1024, 0, 0);
    }
    const float pv[4] = {p4.x, p4.y, p4.z, p4.w};
    const float tv[4] = {t4.x, t4.y, t4.z, t4.w};
    const float mv[4] = {m4.x, m4.y, m4.z, m4.w};
    const float rv[4] = {r4.x, r4.y, r4.z, r4.w};
#pragma unroll
    for (int e = 0; e < 4; ++e) {
      bool pos, neg;
      const float l = loss_elem(pv[e], tv[e], mv[e], pos, neg);
      if (pos) { posCnt++; posSum += l; }
      if (neg) {
        negCnt++;
        atomicAdd(&h[__float_as_uint(rv[e]) >> 16], 1u);
      }
    }
  }

  __shared__ float    sf[256];
  __shared__ uint32_t si[256];
  sf[tid] = posSum;  __syncthreads();
  for (int s = 128; s > 0; s >>= 1) { if (tid < s) sf[tid] += sf[tid + s]; __syncthreads(); }
  if (tid == 0) posPartials[(size_t)b * kBlkPerSample + blockIdx.x] = sf[0];

  si[tid] = posCnt;  __syncthreads();
  for (int s = 128; s > 0; s >>= 1) { if (tid < s) si[tid] += si[tid + s]; __syncthreads(); }
  if (tid == 0) atomicAdd(&numPos[b], si[0]);
  __syncthreads();
  si[tid] = negCnt;  __syncthreads();
  for (int s = 128; s > 0; s >>= 1) { if (tid < s) si[tid] += si[tid + s]; __syncthreads(); }
  if (tid == 0) atomicAdd(&numNeg[b], si[0]);
}

// ---------------- Resolve A: pick top-16-bit prefix, re-zero hist ----------------
__global__ __launch_bounds__(256)
void det_resolveA_kernel(uint32_t* __restrict__ hist,
                         const uint32_t* __restrict__ numNeg,
                         uint32_t* __restrict__ prefix16,
                         uint32_t* __restrict__ neededB) {
  const int b   = blockIdx.x;
  const int tid = threadIdx.x;
  uint32_t* __restrict__ h = hist + (size_t)b * kHistBins;
  __shared__ uint32_t gsum[256];
  uint32_t acc = 0;
  for (int i = 0; i < 256; ++i) acc += h[tid * 256 + i];
  gsum[tid] = acc;
  __syncthreads();
  if (tid == 0) {
    uint32_t needed = numNeg[b];
    if (needed > (uint32_t)kNumNeg) needed = (uint32_t)kNumNeg;
    if (needed == 0u) {
      prefix16[b] = 0xFFFFFFFFu;     // sentinel: nothing selectable
      neededB[b]  = 0u;
    } else {
      uint32_t cum = 0; int g = 255;
      for (; g > 0; --g) { if (cum + gsum[g] >= needed) break; cum += gsum[g]; }
      int v = 255;
      for (; v > 0; --v) { const uint32_t c = h[g * 256 + v];
                           if (cum + c >= needed) break; cum += c; }
      prefix16[b] = (uint32_t)(g * 256 + v);
      neededB[b]  = needed - cum;
    }
  }
  __syncthreads();
  for (int i = tid; i < kHistBins; i += 256) h[i] = 0u;   // reuse for phase B
}

// ---------------- K2: histogram of low 16 bits within prefix ----------------
__global__ __launch_bounds__(256)
void det_histB_kernel(const float* __restrict__ target,
                      const float* __restrict__ rnd,
                      uint32_t* __restrict__ hist,
                      const uint32_t* __restrict__ prefix16) {
  const int b  = blockIdx.y;
  const uint32_t pf = prefix16[b];
  if (pf > 0xFFFFu) return;
  const int tid = threadIdx.x;
  const size_t sbase = (size_t)b * kP;
  const int chunk = blockIdx.x * 4096;
  uint32_t* __restrict__ h = hist + (size_t)b * kHistBins;
#pragma unroll
  for (int j = 0; j < 4; ++j) {
    const size_t off = sbase + (size_t)chunk + (size_t)j * 1024 + (size_t)tid * 4;
    const float4 t4 = *(const float4*)(target + off);
    const float4 r4 = *(const float4*)(rnd    + off);
    const float tv[4] = {t4.x, t4.y, t4.z, t4.w};
    const float rv[4] = {r4.x, r4.y, r4.z, r4.w};
#pragma unroll
    for (int e = 0; e < 4; ++e) {
      if (tv[e] == 0.0f) {
        const uint32_t key = __float_as_uint(rv[e]);
        if ((key >> 16) == pf) atomicAdd(&h[key & 0xFFFFu], 1u);
      }
    }
  }
}

// ---------------- Resolve B: exact key threshold + tie budget ----------------
__global__ __launch_bounds__(256)
void det_resolveB_kernel(const uint32_t* __restrict__ hist,
                         const uint32_t* __restrict__ prefix16,
                         const uint32_t* __restrict__ neededB,
                         uint32_t* __restrict__ Kkey,
                         int*      __restrict__ tieBudget,
                         uint32_t* __restrict__ selIdx) {
  const int b   = blockIdx.x;
  const int tid = threadIdx.x;
  const uint32_t* __restrict__ h = hist + (size_t)b * kHistBins;
  __shared__ uint32_t gsum[256];
  uint32_t acc = 0;
  for (int i = 0; i < 256; ++i) acc += h[tid * 256 + i];
  gsum[tid] = acc;
  __syncthreads();
  if (tid == 0) {
    const uint32_t pf = prefix16[b];
    const uint32_t nd = neededB[b];
    if (pf > 0xFFFFu || nd == 0u) {
      Kkey[b] = 0xFFFFFFFFu;     // select nothing
      tieBudget[b] = 0;
    } else {
      uint32_t cum = 0; int g = 255;
      for (; g > 0; --g) { if (cum + gsum[g] >= nd) break; cum += gsum[g]; }
      int v = 255;
      for (; v > 0; --v) { const uint32_t c = h[g * 256 + v];
                           if (cum + c >= nd) break; cum += c; }
      Kkey[b]      = (pf << 16) | (uint32_t)(g * 256 + v);
      tieBudget[b] = (int)(nd - cum);
    }
    selIdx[b] = 0u;
  }
}

// ---------------- K3: compact selected negatives' losses ----------------
__global__ __launch_bounds__(256)
void det_compact_kernel(const float* __restrict__ pred,
                        const float* __restrict__ target,
                        const float* __restrict__ mask,
                        const float* __restrict__ rnd,
                        const uint32_t* __restrict__ Kkey,
                        int*      __restrict__ tieBudget,
                        uint32_t* __restrict__ selIdx,
                        float*    __restrict__ selLoss) {
  const int b   = blockIdx.y;
  const int tid = threadIdx.x;
  const uint32_t K = Kkey[b];
  const size_t sbase = (size_t)b * kP;
  const int chunk = blockIdx.x * 4096;
#pragma unroll
  for (int j = 0; j < 4; ++j) {
    const size_t off = sbase + (size_t)chunk + (size_t)j * 1024 + (size_t)tid * 4;
    const float4 p4 = *(const float4*)(pred   + off);
    const float4 t4 = *(const float4*)(target + off);
    const float4 m4 = *(const float4*)(mask   + off);
    const float4 r4 = *(const float4*)(rnd    + off);
    if (j < 3) __builtin_prefetch(pred + off + 1024, 0, 0);
    const float pv[4] = {p4.x, p4.y, p4.z, p4.w};
    const float tv[4] = {t4.x, t4.y, t4.z, t4.w};
    const float mv[4] = {m4.x, m4.y, m4.z, m4.w};
    const float rv[4] = {r4.x, r4.y, r4.z, r4.w};
#pragma unroll
    for (int e = 0; e < 4; ++e) {
      bool pos, neg;
      const float l = loss_elem(pv[e], tv[e], mv[e], pos, neg);
      if (neg) {
        const uint32_t key = __float_as_uint(rv[e]);
        bool sel = false;
        if (key > K) sel = true;
        else if (key == K) { if (atomicSub(&tieBudget[b], 1) > 0) sel = true; }
        if (sel) {
          const uint32_t idx = atomicAdd(&selIdx[b], 1u);
          if (idx < (uint32_t)kNumNeg) selLoss[(size_t)b * kNumNeg + idx] = l;
        }
      }
    }
  }
}

// ---------------- K4: per-sample finalize (LDS radix select + sums) -------
__global__ __launch_bounds__(1024)
void det_final_kernel(const float* __restrict__ selLoss,
                      const uint32_t* __restrict__ selIdx,
                      const uint32_t* __restrict__ numPos,
                      const float* __restrict__ posPartials,
                      float* __restrict__ perSample) {
  const int b   = blockIdx.x;
  const int tid = threadIdx.x;
  __shared__ float    sl[kNumNeg];
  __shared__ uint32_t shist[256];
  __shared__ float    sred[1024];
  __shared__ uint32_t sPrefix, sNeeded;

  int n = (int)selIdx[b];
  if (n > kNumNeg) n = kNumNeg;
  const float* __restrict__ g = selLoss + (size_t)b * kNumNeg;

  // --- bring the selected losses into LDS: async DMA path on CDNA5 ---
#ifdef HAVE_ASYNC_LDS
  for (int i = tid; i < n; i += 1024) {
    __builtin_amdgcn_global_load_async_to_lds_b32(
        (gbl_i32_t*)(g + i), (lds_i32_t*)(&sl[i]), 0, 0);
  }
#if __has_builtin(__builtin_amdgcn_s_wait_asynccnt)
  __builtin_amdgcn_s_wait_asynccnt(0);
#else
  asm volatile("s_wait_asynccnt 0" ::: "memory");
#endif
  __syncthreads();
#else
  for (int i = tid; i < n; i += 1024) sl[i] = g[i];
  __syncthreads();
#endif

  const int np = (int)numPos[b];
  int k;
  if (np > 0) { k = 100 * np; if (k > n) k = n; }
  else        { k = (n < 100) ? n : 100; }

  float negSum = 0.0f;
  if (k >= n) {
    float a = 0.0f;
    for (int i = tid; i < n; i += 1024) a += sl[i];
    negSum = block_reduce_1024(a, sred, tid);     // WMMA wave sums
  } else {
    // radix-select the k-th largest loss (losses are >= 0 -> bits monotonic)
    uint32_t prefix = 0u, needed = (uint32_t)k;
    for (int pass = 0; pass < 4; ++pass) {
      const int shift = 24 - pass * 8;
      const uint32_t hiMask = (pass == 0) ? 0u : (0xFFFFFFFFu << (shift + 8));
      for (int i = tid; i < 256; i += 1024) shist[i] = 0u;
      __syncthreads();
      for (int i = tid; i < n; i += 1024) {
        const uint32_t key = __float_as_uint(sl[i]);
        if ((key & hiMask) == prefix) atomicAdd(&shist[(key >> shift) & 0xFFu], 1u);
      }
      __syncthreads();
      if (tid == 0) {
        uint32_t cum = 0; int v = 255;
        for (; v > 0; --v) { const uint32_t c = shist[v];
                             if (cum + c >= needed) break; cum += c; }
        sPrefix = prefix | ((uint32_t)v << shift);
        sNeeded = needed - cum;
      }
      __syncthreads();
      prefix = sPrefix; needed = sNeeded;
      __syncthreads();
    }
    const uint32_t K   = prefix;
    const float    kvl = __uint_as_float(K);
    float a = 0.0f;
    for (int i = tid; i < n; i += 1024)
      if (__float_as_uint(sl[i]) > K) a += sl[i];
    const float S = block_reduce_1024(a, sred, tid);  // WMMA wave sums
    negSum = S + (float)needed * kvl;   // ties contribute identical values
  }

  // positive partials (64 deterministic per-block sums)
  float pv = (tid < kBlkPerSample) ? posPartials[(size_t)b * kBlkPerSample + tid] : 0.0f;
  const float posSum = block_reduce_1024(pv, sred, tid);

  if (tid == 0) {
    const float posL  = (np > 0) ? posSum / (float)np : 0.0f;
    const float denom = (np > 0) ? (float)np : 1.0f;
    perSample[2 * b]     = posL;
    perSample[2 * b + 1] = negSum / denom;
  }
}

// ---------------- K5: combine across batch (fixed order) ----------------
__global__ void det_out_kernel(const float* __restrict__ perSample,
                               float* __restrict__ out) {
  if (blockIdx.x == 0 && threadIdx.x == 0) {
    float sp = 0.0f, sn = 0.0f;
    for (int b = 0; b < kB; ++b) { sp += perSample[2 * b]; sn += perSample[2 * b + 1]; }
    out[0] = sp / (float)kB;
    out[1] = sn / (float)kB;
  }
}

// ---------------- host launcher ----------------
extern "C" void kernel_launch(void* const* d_in, const int* in_sizes, int n_in,
                              void* d_out, int out_size, void* d_ws, size_t ws_size,
                              hipStream_t stream) {
  const float* pred   = (const float*)d_in[0];
  const float* target = (const float*)d_in[1];
  const float* mask   = (const float*)d_in[2];
  const float* rnd    = (const float*)d_in[3];
  float* out = (float*)d_out;

  // workspace layout (all 4-byte words)
  uint32_t* w        = (uint32_t*)d_ws;
  uint32_t* hist     = w;                                  // 32 * 65536
  uint32_t* numPos   = hist + (size_t)kB * kHistBins;      // 32
  uint32_t* numNeg   = numPos + kB;                        // 32
  uint32_t* selIdx   = numNeg + kB;                        // 32
  int*      tieBud   = (int*)(selIdx + kB);                // 32
  uint32_t* prefix16 = (uint32_t*)(tieBud + kB);           // 32
  uint32_t* neededB  = prefix16 + kB;                      // 32
  uint32_t* Kkey     = neededB + kB;                       // 32
  float*    posPart  = (float*)(Kkey + kB);                // 32 * 64
  float*    perSamp  = posPart + (size_t)kB * kBlkPerSample; // 64
  float*    selLoss  = perSamp + 2 * kB;                   // 32 * 10000

  const int zeroWords = kB * kHistBins + 7 * kB + kB * kBlkPerSample + 2 * kB;

  det_zero_kernel<<<(zeroWords + 255) / 256, 256, 0, stream>>>(w, zeroWords);

  dim3 gridMain(kBlkPerSample, kB);
  det_main_kernel<<<gridMain, 256, 0, stream>>>(pred, target, mask, rnd,
                                                hist, numPos, numNeg, posPart);
  det_resolveA_kernel<<<kB, 256, 0, stream>>>(hist, numNeg, prefix16, neededB);
  det_histB_kernel<<<gridMain, 256, 0, stream>>>(target, rnd, hist, prefix16);
  det_resolveB_kernel<<<kB, 256, 0, stream>>>(hist, prefix16, neededB,
                                              Kkey, tieBud, selIdx);
  det_compact_kernel<<<gridMain, 256, 0, stream>>>(pred, target, mask, rnd,
                                                   Kkey, tieBud, selIdx, selLoss);
  det_final_kernel<<<kB, 1024, 0, stream>>>(selLoss, selIdx, numPos, posPart, perSamp);
  det_out_kernel<<<1, 32, 0, stream>>>(perSamp, out);

  (void)in_sizes; (void)n_in; (void)out_size; (void)ws_size;
}